// MPNNModel_5153960755353
// MI455X (gfx1250) — compile-verified
//
#include <hip/hip_runtime.h>

#define N_NODES 50000
#define N_EDGES 800000
#define IN_DIM  64
#define HID_DIM 128
#define OUT_DIM 32

typedef __attribute__((ext_vector_type(2))) float v2f;
typedef __attribute__((ext_vector_type(8))) float v8f;

// ---------------- zero fill (float4 grid-stride) ----------------
__global__ void zero_f4_kernel(float4* __restrict__ p, int n4) {
    int i = blockIdx.x * blockDim.x + threadIdx.x;
    int stride = gridDim.x * blockDim.x;
    float4 z; z.x = 0.f; z.y = 0.f; z.z = 0.f; z.w = 0.f;
    for (; i < n4; i += stride) p[i] = z;
}

// ---------------- edge scatter: agg[dst] += feat[src]; deg[dst] += 1 ----------------
// One wave (32 lanes) per edge iteration; lane covers DIM/32 elements.
template<int DIM>
__global__ void scatter_kernel(const float* __restrict__ feat,
                               const long long* __restrict__ src,
                               const long long* __restrict__ dst,
                               float* __restrict__ agg,
                               float* __restrict__ deg,   // nullptr -> skip degree count
                               int n_edges) {
    const int lane  = threadIdx.x & 31;
    int wave        = (blockIdx.x * blockDim.x + threadIdx.x) >> 5;
    const int nwave = (gridDim.x * blockDim.x) >> 5;
    for (int e = wave; e < n_edges; e += nwave) {
        const long long s = src[e];
        const long long d = dst[e];
        const float* fs = feat + s * (long long)DIM;
        float*       ad = agg  + d * (long long)DIM;
#pragma unroll
        for (int j = 0; j < DIM; j += 32) {
            unsafeAtomicAdd(&ad[j + lane], fs[j + lane]);   // global_atomic_add_f32
        }
        if (deg != nullptr && lane == 0) {
            unsafeAtomicAdd(&deg[d], 1.0f);
        }
    }
}

// ---------------- WMMA fp32 GEMM: out = epilogue(A[M,K] @ W[K,N]) ----------------
// One wave per 16x16 output tile. grid = (M/16, N/16), block = 32 (EXEC all ones).
// MEAN_MASK_RELU: out = deg>0 ? relu(acc/deg + bias) : 0   (mean-aggregate layer)
// else:           out = acc + bias                          (output projection)
template<int K, int N, bool MEAN_MASK_RELU>
__global__ void wmma_gemm_kernel(const float* __restrict__ A,
                                 const float* __restrict__ W,
                                 const float* __restrict__ bias,
                                 const float* __restrict__ deg,
                                 float* __restrict__ out) {
    const int lane = threadIdx.x & 31;
    const int m0 = blockIdx.x * 16;
    const int n0 = blockIdx.y * 16;

    // A 16x4 f32 layout: lanes 0-15 hold row M=lane, K = k0,k0+1 in vgpr0/1;
    // lanes 16-31 hold row M=lane-16, K = k0+2,k0+3.
    const int row = lane & 15;                 // M row (A) / N col (B)
    const int kh  = (lane >> 4) << 1;          // 0 or 2

    const float* Arow = A + (long long)(m0 + row) * K;
    const float* Wcol = W + (n0 + row);

    v8f acc = {};
#pragma unroll
    for (int kk = 0; kk < K; kk += 4) {
        v2f a, b;
        a.x = Arow[kk + kh];
        a.y = Arow[kk + kh + 1];
        b.x = Wcol[(kk + kh) * N];
        b.y = Wcol[(kk + kh + 1) * N];
        acc = __builtin_amdgcn_wmma_f32_16x16x4_f32(
            /*neg_a=*/false, a, /*neg_b=*/false, b,
            /*c_mod=*/(short)0, acc, /*reuse_a=*/false, /*reuse_b=*/false);
    }

    // C/D 16x16 f32 layout: vgpr r -> M = r + 8*(lane>=16), N = lane&15
    const int moff = (lane >> 4) << 3;         // 0 or 8
    const int ncol = lane & 15;
    const float bn = bias[n0 + ncol];
#pragma unroll
    for (int r = 0; r < 8; ++r) {
        const int m = m0 + moff + r;
        float v = acc[r];
        if constexpr (MEAN_MASK_RELU) {
            const float d = deg[m];
            v = (d > 0.0f) ? fmaxf(v / d + bn, 0.0f) : 0.0f;
        } else {
            v = v + bn;
        }
        out[(long long)m * N + n0 + ncol] = v;
    }
}

extern "C" void kernel_launch(void* const* d_in, const int* in_sizes, int n_in,
                              void* d_out, int out_size, void* d_ws, size_t ws_size,
                              hipStream_t stream) {
    const float*     x    = (const float*)d_in[0];
    const long long* ei   = (const long long*)d_in[1];   // int64 [2, E]
    const float*     W1   = (const float*)d_in[2];
    const float*     b1   = (const float*)d_in[3];
    const float*     W2   = (const float*)d_in[4];
    const float*     b2   = (const float*)d_in[5];
    const float*     Wout = (const float*)d_in[6];
    const float*     bout = (const float*)d_in[7];
    float*           out  = (float*)d_out;

    const long long* src = ei;
    const long long* dst = ei + N_EDGES;

    // Workspace layout (floats): [deg pad 50048][agg 50000*128][h 50000*128]
    const int DEG_PAD = 50048;                       // 64-aligned, 16B-aligned offset
    float* deg = (float*)d_ws;
    float* agg = deg + DEG_PAD;                      // layer1 uses stride 64 (first half)
    float* h   = agg + (long long)N_NODES * HID_DIM; // h1, then overwritten by h2

    const int MT = N_NODES / 16;                     // 3125, exact

    // ---- Layer 1 ----
    {
        int n4 = (DEG_PAD + N_NODES * IN_DIM) / 4;   // deg + agg1 region
        zero_f4_kernel<<<2048, 256, 0, stream>>>((float4*)d_ws, n4);
        scatter_kernel<IN_DIM><<<1024, 256, 0, stream>>>(x, src, dst, agg, deg, N_EDGES);
        wmma_gemm_kernel<IN_DIM, HID_DIM, true>
            <<<dim3(MT, HID_DIM / 16), 32, 0, stream>>>(agg, W1, b1, deg, h);
    }
    // ---- Layer 2 ----
    {
        int n4 = (N_NODES * HID_DIM) / 4;
        zero_f4_kernel<<<2048, 256, 0, stream>>>((float4*)agg, n4);
        scatter_kernel<HID_DIM><<<1024, 256, 0, stream>>>(h, src, dst, agg, nullptr, N_EDGES);
        wmma_gemm_kernel<HID_DIM, HID_DIM, true>
            <<<dim3(MT, HID_DIM / 16), 32, 0, stream>>>(agg, W2, b2, deg, h);
    }
    // ---- Output projection: out = h2 @ Wout + bout ----
    wmma_gemm_kernel<HID_DIM, OUT_DIM, false>
        <<<dim3(MT, OUT_DIM / 16), 32, 0, stream>>>(h, Wout, bout, nullptr, out);
}